// SpatialAttention_40544491274439
// MI455X (gfx1250) — compile-verified
//
#include <hip/hip_runtime.h>
#include <math.h>

#define Hh 256
#define Ww 256
#define Bb 8
#define Cc 64
#define HW (Hh * Ww)
#define NPIX (Bb * HW)          // 524288
#define K2_BLOCKS (NPIX / 128)  // 4096 blocks, 128 pixels per 256-thread block
#define EPSV 1e-5f

typedef __attribute__((ext_vector_type(2))) float v2f;
typedef __attribute__((ext_vector_type(8))) float v8f;

// ---------------------------------------------------------------------------
// Kernel 1: feat[b,y,x] = (mean_c x, max_c x), interleaved float2.
// Coalesced: lanes cover consecutive x; 64 strided channel reads each.
// ---------------------------------------------------------------------------
__global__ __launch_bounds__(256) void k1_reduce(const float* __restrict__ x,
                                                 float2* __restrict__ feat) {
    const int pid = blockIdx.x * 256 + threadIdx.x;   // < NPIX
    const int b = pid / HW;
    const int s = pid - b * HW;
    const float* p = x + (size_t)b * Cc * HW + s;
    float sum = 0.f;
    float mx = -INFINITY;
#pragma unroll 8
    for (int c = 0; c < Cc; ++c) {
        const float v = p[(size_t)c * HW];
        sum += v;
        mx = fmaxf(mx, v);
    }
    feat[pid] = make_float2(sum * (1.0f / Cc), mx);
}

// ---------------------------------------------------------------------------
// Kernel 2: deformable 3x3 conv (Cin=2, Cout=1) via chained f32 WMMAs.
// Wave = 16 pixels (M). K ordered t' = 2k+ci, padded 18 -> 20 (5 chunks of 4).
// Lane (half = lane>>4) computes tap k = 2c+half per chunk: one bilinear
// position feeds both channels (A = {s_avg, s_max}, B = {w[k], w[9+k]}).
// feat (4.2 MB) is L2-resident -> gathers are cheap.
// ---------------------------------------------------------------------------
__global__ __launch_bounds__(256) void k2_deform(const float* __restrict__ off,
                                                 const float* __restrict__ wgt,
                                                 const float2* __restrict__ feat,
                                                 float* __restrict__ out,
                                                 float2* __restrict__ partials) {
    const int tid = threadIdx.x;
    const int lane = tid & 31;
    const int wv = tid >> 5;
    const int pixbase = blockIdx.x * 128 + wv * 16;   // 16 contiguous pixels in a row
    const int b = pixbase / HW;
    const int rem = pixbase - b * HW;
    const int y = rem >> 8;          // W = 256
    const int x0 = rem & 255;
    const int half = lane >> 4;
    const int M = lane & 15;
    const int x = x0 + M;

    const float* offp = off + (size_t)b * 18 * HW + y * Ww + x;
    const float2* fb = feat + (size_t)b * HW;

    v8f acc = {0.f, 0.f, 0.f, 0.f, 0.f, 0.f, 0.f, 0.f};

#pragma unroll
    for (int c = 0; c < 5; ++c) {
        const int k = 2 * c + half;          // tap index for this lane
        float s0 = 0.f, s1 = 0.f, w0 = 0.f, w1 = 0.f;
        if (k < 9) {                          // k==9 (chunk 4, half 1) is zero pad
            const float dy = offp[(size_t)(2 * k) * HW];
            const float dx = offp[(size_t)(2 * k + 1) * HW];
            const float py = (float)(y - 1 + (k / 3)) + dy;
            const float px = (float)(x - 1 + (k % 3)) + dx;
            const float fy = floorf(py), fxx = floorf(px);
            const int iy0 = (int)fy, ix0 = (int)fxx;
            const int iy1 = iy0 + 1, ix1 = ix0 + 1;
            const float ry = py - fy, rx = px - fxx;
            const float w00 = (1.f - ry) * (1.f - rx);
            const float w01 = (1.f - ry) * rx;
            const float w10 = ry * (1.f - rx);
            const float w11 = ry * rx;
            auto corner = [&](int iy, int ix, float wq) {
                const bool v = ((unsigned)iy < (unsigned)Hh) & ((unsigned)ix < (unsigned)Ww);
                const int iyc = min(max(iy, 0), Hh - 1);
                const int ixc = min(max(ix, 0), Ww - 1);
                const float2 f = fb[iyc * Ww + ixc];
                const float m = v ? wq : 0.f;
                s0 = fmaf(m, f.x, s0);
                s1 = fmaf(m, f.y, s1);
            };
            corner(iy0, ix0, w00);
            corner(iy0, ix1, w01);
            corner(iy1, ix0, w10);
            corner(iy1, ix1, w11);
            w0 = wgt[k];        // ci = 0 (avg channel)
            w1 = wgt[9 + k];    // ci = 1 (max channel)
        }
        v2f a;  a[0] = s0;  a[1] = s1;   // A: M = lane&15, K = 4c + 2*half + {0,1}
        v2f bv; bv[0] = w0; bv[1] = w1;  // B: identical across all N columns
        // EXEC is all-ones here (divergence only inside the guarded block above)
        acc = __builtin_amdgcn_wmma_f32_16x16x4_f32(false, a, false, bv,
                                                    (short)0, acc, false, false);
    }

    // C/D layout: VGPR r, lanes 0-15 -> M=r ; lanes 16-31 -> M=8+r.
    // Lane l reads acc[l&7] => pixel p = (l&7) + 8*(l>>4); lanes with !(l&8) write.
    const int r = lane & 7;
    const float e0 = (r & 1) ? acc[1] : acc[0];
    const float e1 = (r & 1) ? acc[3] : acc[2];
    const float e2 = (r & 1) ? acc[5] : acc[4];
    const float e3 = (r & 1) ? acc[7] : acc[6];
    const float f0 = (r & 2) ? e1 : e0;
    const float f1 = (r & 2) ? e3 : e2;
    const float val = (r & 4) ? f1 : f0;
    const bool writer = (lane & 8) == 0;
    const int p = r + (half << 3);
    if (writer) out[pixbase + p] = val;

    // Deterministic block-level (sum, sumsq) -> partials (no atomics)
    __shared__ float ssum[256];
    __shared__ float ssq[256];
    ssum[tid] = writer ? val : 0.f;
    ssq[tid]  = writer ? val * val : 0.f;
    for (int sft = 128; sft > 0; sft >>= 1) {
        __syncthreads();
        if (tid < sft) {
            ssum[tid] += ssum[tid + sft];
            ssq[tid]  += ssq[tid + sft];
        }
    }
    if (tid == 0) partials[blockIdx.x] = make_float2(ssum[0], ssq[0]);
}

// ---------------------------------------------------------------------------
// Kernel 2b: single block, fixed-order reduction of 4096 partials -> stats
// stats[0] = mean, stats[1] = rsqrt(var + eps)
// ---------------------------------------------------------------------------
__global__ __launch_bounds__(256) void k2b_stats(const float2* __restrict__ partials,
                                                 float* __restrict__ stats) {
    const int tid = threadIdx.x;
    __shared__ float ssum[256];
    __shared__ float ssq[256];
    float a = 0.f, q = 0.f;
    for (int i = tid; i < K2_BLOCKS; i += 256) {
        const float2 pv = partials[i];
        a += pv.x;
        q += pv.y;
    }
    ssum[tid] = a;
    ssq[tid] = q;
    for (int sft = 128; sft > 0; sft >>= 1) {
        __syncthreads();
        if (tid < sft) {
            ssum[tid] += ssum[tid + sft];
            ssq[tid]  += ssq[tid + sft];
        }
    }
    if (tid == 0) {
        const float mean = ssum[0] * (1.0f / (float)NPIX);
        const float var = ssq[0] * (1.0f / (float)NPIX) - mean * mean;
        stats[0] = mean;
        stats[1] = rsqrtf(var + EPSV);
    }
}

// ---------------------------------------------------------------------------
// Kernel 3: y = x * sigmoid(gamma*(out-mu)*inv + beta), float4 vectorized.
// out (2 MB) is L2-resident; x read/write is the HBM bound.
// ---------------------------------------------------------------------------
__global__ __launch_bounds__(256) void k3_apply(const float4* __restrict__ x4,
                                                const float* __restrict__ out,
                                                const float* __restrict__ stats,
                                                const float* __restrict__ gamma,
                                                const float* __restrict__ beta,
                                                float4* __restrict__ y4) {
    const long i4 = (long)blockIdx.x * 256 + threadIdx.x;   // < NPIX*Cc/4
    const long e = i4 * 4;
    const long b = e / ((long)Cc * HW);
    const long s = e % (long)HW;                            // spatial within image
    const float4 xv = x4[i4];
    const float4 ov = *(const float4*)(out + b * HW + s);
    const float mean = stats[0];
    const float inv = stats[1];
    const float g = gamma[0];
    const float be = beta[0];
    float4 r;
    {
        const float t0 = fmaf(g * (ov.x - mean), inv, be);
        const float t1 = fmaf(g * (ov.y - mean), inv, be);
        const float t2 = fmaf(g * (ov.z - mean), inv, be);
        const float t3 = fmaf(g * (ov.w - mean), inv, be);
        r.x = xv.x / (1.f + expf(-t0));
        r.y = xv.y / (1.f + expf(-t1));
        r.z = xv.z / (1.f + expf(-t2));
        r.w = xv.w / (1.f + expf(-t3));
    }
    y4[i4] = r;
}

// ---------------------------------------------------------------------------
extern "C" void kernel_launch(void* const* d_in, const int* in_sizes, int n_in,
                              void* d_out, int out_size, void* d_ws, size_t ws_size,
                              hipStream_t stream) {
    const float* x      = (const float*)d_in[0];   // [8,64,256,256]
    const float* offset = (const float*)d_in[1];   // [8,18,256,256]
    const float* weight = (const float*)d_in[2];   // [1,2,3,3]
    const float* gamma  = (const float*)d_in[3];   // [1]
    const float* beta   = (const float*)d_in[4];   // [1]
    float* y = (float*)d_out;                      // [8,64,256,256]

    // Workspace layout (all 16B-aligned): feat | out | partials | stats
    char* ws = (char*)d_ws;
    float2* feat     = (float2*)ws;                               // NPIX * 8 B
    float*  convout  = (float*)(ws + (size_t)NPIX * 8);           // NPIX * 4 B
    float2* partials = (float2*)(ws + (size_t)NPIX * 12);         // 4096 * 8 B
    float*  stats    = (float*)(ws + (size_t)NPIX * 12 + K2_BLOCKS * 8);

    k1_reduce<<<NPIX / 256, 256, 0, stream>>>(x, feat);
    k2_deform<<<K2_BLOCKS, 256, 0, stream>>>(offset, weight, feat, convout, partials);
    k2b_stats<<<1, 256, 0, stream>>>(partials, stats);
    k3_apply<<<(int)(((long)NPIX * Cc / 4) / 256), 256, 0, stream>>>(
        (const float4*)x, convout, stats, gamma, beta, (float4*)y);
}